// DecayTGNMemoryModule_69063074120439
// MI455X (gfx1250) — compile-verified
//
#include <hip/hip_runtime.h>
#include <hip/hip_bf16.h>
#include <math.h>
#include <stdint.h>

typedef float v2f __attribute__((ext_vector_type(2)));
typedef float v8f __attribute__((ext_vector_type(8)));

#define NUM_NODES 200000
#define MEM_DIM   128
#define MSG_DIM   172
#define BATCH     65536
#define DECAY     0.1f
#define TSTR      132   // padded LDS tile stride (132 mod 64 == 4 -> conflict-free fragments)

__device__ __forceinline__ v8f wmma_f32(v8f acc, v2f a, v2f b) {
  // D = A(16x4,f32) * B(4x16,f32) + C ; exact fp32, RNE
  return __builtin_amdgcn_wmma_f32_16x16x4_f32(false, a, false, b,
                                               (short)0, acc, false, false);
}

// CDNA5 async copy: global -> LDS, 16B per lane, tracked by ASYNCcnt.
// vdst operand is the LDS *byte offset* (low 32 bits of the flat shared addr).
__device__ __forceinline__ void async_g2l_b128(void* lds_ptr, const void* gptr) {
  uint32_t lds_off = (uint32_t)(uintptr_t)lds_ptr;
  asm volatile("global_load_async_to_lds_b128 %0, %1, off"
               :: "v"(lds_off), "v"((uint64_t)(uintptr_t)gptr) : "memory");
}
__device__ __forceinline__ void wait_async0() {
  asm volatile("s_wait_asynccnt 0" ::: "memory");
}

__global__ void k_init(int* last_pos) {
  int i = blockIdx.x * 256 + threadIdx.x;
  if (i < NUM_NODES) last_pos[i] = -1;
}

__global__ void k_copy(const float4* __restrict__ src, float4* __restrict__ dst) {
  int i = blockIdx.x * 256 + threadIdx.x;
  if (i < NUM_NODES * MEM_DIM / 4) dst[i] = src[i];
}

__global__ void k_maxpos(const int* __restrict__ ids, int* last_pos) {
  int i = blockIdx.x * 256 + threadIdx.x;
  if (i < BATCH) {
    int id = ids[i];
    id = id < 0 ? 0 : (id >= NUM_NODES ? NUM_NODES - 1 : id);
    atomicMax(&last_pos[id], i);
  }
}

__global__ __launch_bounds__(64)
void k_gru(const int* __restrict__ node_ids,
           const float* __restrict__ messages,
           const float* __restrict__ ts,
           const float* __restrict__ memory,
           const float* __restrict__ last_update,
           const float* __restrict__ W1, const float* __restrict__ b1,
           const float* __restrict__ W2, const float* __restrict__ b2,
           const float* __restrict__ W_ih, const float* __restrict__ W_hh,
           const float* __restrict__ b_ih, const float* __restrict__ b_hh,
           const int* __restrict__ last_pos,
           float* __restrict__ out)
{
  __shared__ __align__(16) float msgT[2 * 16 * MSG_DIM]; // async-staged message tiles
  __shared__ __align__(16) float t0[2 * 16 * TSTR];      // hidden tile, reused as h tile
  __shared__ __align__(16) float t1[2 * 16 * TSTR];      // pm tile
  __shared__ float sc[2 * 16];
  __shared__ int   idr[2 * 16];
  __shared__ int   win[2 * 16];

  const int w    = threadIdx.x >> 5;
  const int lane = threadIdx.x & 31;
  const int half = lane >> 4;    // 0: K=0,1 / rows 0-7 ; 1: K=2,3 / rows 8-15
  const int n16  = lane & 15;
  const int row0 = (blockIdx.x * 2 + w) * 16;   // 16 events per wave

  float* Tmsg = msgT + w * 16 * MSG_DIM;
  float* Thid = t0 + w * 16 * TSTR;
  float* Tpm  = t1 + w * 16 * TSTR;

  // -------- async stage: 16x172 message tile (contiguous 11008 B) into LDS ------
  {
    const float* gbase = messages + (size_t)row0 * MSG_DIM;
    for (int e = lane; e < 16 * MSG_DIM / 4; e += 32)       // 688 x b128
      async_g2l_b128(Tmsg + e * 4, gbase + e * 4);
    wait_async0();
  }
  __syncthreads();

  // ---------------- GEMM1: hidden = relu(msg @ W1^T + b1), K=172 ----------------
  {
    const float* Ap = Tmsg + n16 * MSG_DIM + 2 * half;
    for (int nt = 0; nt < 8; ++nt) {
      v8f acc = {};
      const float* Wn = W1 + (size_t)(nt * 16 + n16) * MSG_DIM + 2 * half;
      for (int k = 0; k < MSG_DIM; k += 4) {
        v2f a = *(const v2f*)(Ap + k);
        v2f b = *(const v2f*)(Wn + k);
        acc = wmma_f32(acc, a, b);
      }
      float bias = b1[nt * 16 + n16];
#pragma unroll
      for (int r = 0; r < 8; ++r) {
        float v = acc[r] + bias;
        Thid[(r + 8 * half) * TSTR + nt * 16 + n16] = v > 0.f ? v : 0.f;
      }
    }
  }
  __syncthreads();

  // ---------------- GEMM2: pm = hidden @ W2^T + b2, K=128 ----------------
  {
    const float* Ap = Thid + n16 * TSTR + 2 * half;
    for (int nt = 0; nt < 8; ++nt) {
      v8f acc = {};
      const float* Wn = W2 + (size_t)(nt * 16 + n16) * MEM_DIM + 2 * half;
      for (int k = 0; k < MEM_DIM; k += 4) {
        v2f a = *(const v2f*)(Ap + k);
        v2f b = *(const v2f*)(Wn + k);
        acc = wmma_f32(acc, a, b);
      }
      float bias = b2[nt * 16 + n16];
#pragma unroll
      for (int r = 0; r < 8; ++r)
        Tpm[(r + 8 * half) * TSTR + nt * 16 + n16] = acc[r] + bias;
    }
  }
  __syncthreads();

  // ---------------- decay gather: h = memory[ids] * exp(-decay*dt) --------------
  if (lane < 16) {
    int i  = row0 + lane;
    int id = node_ids[i];
    id = id < 0 ? 0 : (id >= NUM_NODES ? NUM_NODES - 1 : id);
    float dt = ts[i] - last_update[id];
    dt = dt > 0.f ? dt : 0.f;
    sc[w * 16 + lane]  = __expf(-DECAY * dt);
    idr[w * 16 + lane] = id;
    win[w * 16 + lane] = (last_pos[id] == i) ? 1 : 0;   // last duplicate wins
  }
  __syncthreads();
  for (int e = lane; e < 16 * MEM_DIM; e += 32) {       // Thid now holds h
    int m = e >> 7, c = e & 127;
    Thid[m * TSTR + c] = memory[(size_t)idr[w * 16 + m] * MEM_DIM + c] * sc[w * 16 + m];
  }
  __syncthreads();

  // ---------- GEMM3/4 + GRU gates; r/z fuse gx+gh into one accumulator ----------
  {
    const float* Apm = Tpm  + n16 * TSTR + 2 * half;
    const float* Ah  = Thid + n16 * TSTR + 2 * half;
    for (int jt = 0; jt < 8; ++jt) {
      int col = jt * 16 + n16;
      v8f ar = {}, az = {}, axn = {}, ahn = {};
      const float* Wr_x = W_ih + (size_t)(col)       * MEM_DIM + 2 * half;
      const float* Wr_h = W_hh + (size_t)(col)       * MEM_DIM + 2 * half;
      const float* Wz_x = W_ih + (size_t)(128 + col) * MEM_DIM + 2 * half;
      const float* Wz_h = W_hh + (size_t)(128 + col) * MEM_DIM + 2 * half;
      const float* Wn_x = W_ih + (size_t)(256 + col) * MEM_DIM + 2 * half;
      const float* Wn_h = W_hh + (size_t)(256 + col) * MEM_DIM + 2 * half;
      for (int k = 0; k < MEM_DIM; k += 4) {
        v2f ax = *(const v2f*)(Apm + k);
        v2f ah = *(const v2f*)(Ah  + k);
        ar  = wmma_f32(ar,  ax, *(const v2f*)(Wr_x + k));
        ar  = wmma_f32(ar,  ah, *(const v2f*)(Wr_h + k));
        az  = wmma_f32(az,  ax, *(const v2f*)(Wz_x + k));
        az  = wmma_f32(az,  ah, *(const v2f*)(Wz_h + k));
        axn = wmma_f32(axn, ax, *(const v2f*)(Wn_x + k));
        ahn = wmma_f32(ahn, ah, *(const v2f*)(Wn_h + k));
      }
      float br  = b_ih[col]       + b_hh[col];
      float bz  = b_ih[128 + col] + b_hh[128 + col];
      float bxn = b_ih[256 + col];
      float bhn = b_hh[256 + col];
#pragma unroll
      for (int r = 0; r < 8; ++r) {
        int   m  = r + 8 * half;
        float rr = 1.f / (1.f + __expf(-(ar[r] + br)));
        float zz = 1.f / (1.f + __expf(-(az[r] + bz)));
        float nn = tanhf(axn[r] + bxn + rr * (ahn[r] + bhn));
        float hv = Thid[m * TSTR + col];
        float hn = (1.f - zz) * nn + zz * hv;
        if (win[w * 16 + m])
          out[(size_t)idr[w * 16 + m] * MEM_DIM + col] = hn;
      }
    }
  }
}

extern "C" void kernel_launch(void* const* d_in, const int* in_sizes, int n_in,
                              void* d_out, int out_size, void* d_ws, size_t ws_size,
                              hipStream_t stream) {
  (void)in_sizes; (void)n_in; (void)out_size; (void)ws_size;
  const int*   node_ids    = (const int*)  d_in[0];
  const float* messages    = (const float*)d_in[1];
  const float* ts          = (const float*)d_in[2];
  const float* memory      = (const float*)d_in[3];
  const float* last_update = (const float*)d_in[4];
  const float* W1   = (const float*)d_in[5];
  const float* b1   = (const float*)d_in[6];
  const float* W2   = (const float*)d_in[7];
  const float* b2   = (const float*)d_in[8];
  const float* W_ih = (const float*)d_in[9];
  const float* W_hh = (const float*)d_in[10];
  const float* b_ih = (const float*)d_in[11];
  const float* b_hh = (const float*)d_in[12];
  float* out = (float*)d_out;
  int* last_pos = (int*)d_ws;   // needs NUM_NODES*4 = 800 KB of ws

  k_init  <<<(NUM_NODES + 255) / 256, 256, 0, stream>>>(last_pos);
  k_copy  <<<(NUM_NODES * MEM_DIM / 4 + 255) / 256, 256, 0, stream>>>(
              (const float4*)memory, (float4*)out);
  k_maxpos<<<(BATCH + 255) / 256, 256, 0, stream>>>(node_ids, last_pos);
  k_gru   <<<BATCH / 32, 64, 0, stream>>>(node_ids, messages, ts, memory, last_update,
              W1, b1, W2, b2, W_ih, W_hh, b_ih, b_hh, last_pos, out);
}